// AdaCoFSampler_74835510166119
// MI455X (gfx1250) — compile-verified
//
#include <hip/hip_runtime.h>
#include <stdint.h>

// AdaCoF deformable sampling for MI455X (gfx1250, wave32).
// B=4, C=3, H=W=256, K2=49 (fixed by the reference).
constexpr int B_ = 4, C_ = 3, H_ = 256, W_ = 256;
constexpr int HW = H_ * W_;
constexpr int K2 = 49;
constexpr int BLOCK = 256;

static_assert(HW * 4 == 262144, "imm offset below assumes HW*4 == 262144");

// ---------------------------------------------------------------------------
// Async global->LDS staging of one tap's streaming operands (dx, dy, weight).
// GVS addressing: saddr = 64-bit base (SGPR pair), vaddr = 32-bit byte offset.
// dy lives exactly HW*4 = 262144 bytes past dx in the offsets tensor, which
// fits the signed 24-bit instruction offset, so dx/dy share one vaddr.
// ---------------------------------------------------------------------------
__device__ __forceinline__ void issue_tap(uint32_t lds_dx, uint32_t lds_dy,
                                          uint32_t lds_wt, uint32_t off_o,
                                          uint32_t off_w,
                                          const float* __restrict__ offs,
                                          const float* __restrict__ wts) {
  asm volatile("global_load_async_to_lds_b32 %0, %1, %2 offset:0"
               :: "v"(lds_dx), "v"(off_o), "s"(offs) : "memory");
  asm volatile("global_load_async_to_lds_b32 %0, %1, %2 offset:262144"
               :: "v"(lds_dy), "v"(off_o), "s"(offs) : "memory");
  asm volatile("global_load_async_to_lds_b32 %0, %1, %2 offset:0"
               :: "v"(lds_wt), "v"(off_w), "s"(wts) : "memory");
}

// ---------------------------------------------------------------------------
// Pre-pass: NCHW -> NHWC(float4, w=0) so each bilinear corner is one B128 load.
// ---------------------------------------------------------------------------
__global__ __launch_bounds__(BLOCK) void nchw_to_nhwc4_kernel(
    const float* __restrict__ img, float4* __restrict__ timg) {
  const int pg = blockIdx.x * BLOCK + threadIdx.x;  // global pixel (b*HW + p)
  const int b = pg / HW;
  const int p = pg - b * HW;
  const float* base = img + (size_t)b * C_ * HW + p;
  float4 v;
  v.x = base[0];
  v.y = base[HW];
  v.z = base[2 * HW];
  v.w = 0.0f;
  timg[pg] = v;
}

// ---------------------------------------------------------------------------
// Main kernel: one thread per (b, y, x). 49 taps, double-buffered async
// staging of (dx, dy, wt) through LDS, bilinear gather from L2-resident image.
// ---------------------------------------------------------------------------
template <bool NHWC>
__global__ __launch_bounds__(BLOCK) void adacof_kernel(
    const float* __restrict__ img, const float* __restrict__ offs,
    const float* __restrict__ wts, const float4* __restrict__ timg,
    float* __restrict__ out) {
  __shared__ float s_stage[2][3][BLOCK];

  const int tid = threadIdx.x;
  const int pg = blockIdx.x * BLOCK + tid;
  const int b = pg / HW;
  const int p = pg - b * HW;
  const int y = p >> 8;
  const int x = p & (W_ - 1);

  // LDS byte offsets for both buffers (low 32 bits of the flat address are
  // the wave-relative LDS offset on CDNA5).
  uint32_t l_dx[2], l_dy[2], l_wt[2];
#pragma unroll
  for (int u = 0; u < 2; ++u) {
    l_dx[u] = (uint32_t)(uintptr_t)&s_stage[u][0][tid];
    l_dy[u] = (uint32_t)(uintptr_t)&s_stage[u][1][tid];
    l_wt[u] = (uint32_t)(uintptr_t)&s_stage[u][2][tid];
  }

  // Byte offsets of tap 0 within the offsets / weights tensors.
  const uint32_t off_o0 = (uint32_t)(((b * 2 * K2) * HW + p) * 4);
  const uint32_t off_w0 = (uint32_t)(((b * K2) * HW + p) * 4);

  // Prologue: stage tap 0 into buffer 0.
  issue_tap(l_dx[0], l_dy[0], l_wt[0], off_o0, off_w0, offs, wts);

  constexpr float SX = (float)W_ / (float)(W_ - 1);
  constexpr float SY = (float)H_ / (float)(H_ - 1);
  const float fx = (float)x, fy = (float)y;

  float acc0 = 0.0f, acc1 = 0.0f, acc2 = 0.0f;
  int buf = 0;

  for (int k = 0; k < K2; ++k) {
    if (k + 1 < K2) {
      const uint32_t no = off_o0 + (uint32_t)(k + 1) * (2u * HW * 4u);
      const uint32_t nw = off_w0 + (uint32_t)(k + 1) * (HW * 4u);
      // WAR guard: tap (k-1)'s ds_loads from buf^1 must be in VGPRs before
      // the async engine overwrites that buffer.
      asm volatile("s_wait_dscnt 0x0" ::: "memory");
      issue_tap(l_dx[buf ^ 1], l_dy[buf ^ 1], l_wt[buf ^ 1], no, nw, offs, wts);
      // Wait for the 3 oldest async loads (tap k) to land in LDS; keep the
      // 3 newest (tap k+1) in flight.
      asm volatile("s_wait_asynccnt 0x3" ::: "memory");
    } else {
      asm volatile("s_wait_asynccnt 0x0" ::: "memory");
    }

    const float dx = s_stage[buf][0][tid];
    const float dy = s_stage[buf][1][tid];
    const float wt = s_stage[buf][2][tid];

    float px = (fx + dx) * SX - 0.5f;
    float py = (fy + dy) * SY - 0.5f;
    px = fminf(fmaxf(px, 0.0f), (float)(W_ - 1));
    py = fminf(fmaxf(py, 0.0f), (float)(H_ - 1));
    const float x0f = floorf(px);
    const float y0f = floorf(py);
    const float wx = px - x0f;
    const float wy = py - y0f;
    const int x0 = (int)x0f;
    const int y0 = (int)y0f;
    const int x1 = min(x0 + 1, W_ - 1);
    const int y1 = min(y0 + 1, H_ - 1);

    const float omx = 1.0f - wx, omy = 1.0f - wy;
    const float w00 = omx * omy * wt;
    const float w01 = wx * omy * wt;
    const float w10 = omx * wy * wt;
    const float w11 = wx * wy * wt;

    if (NHWC) {
      const float4* tb = timg + (size_t)b * HW;
      const float4 v00 = tb[y0 * W_ + x0];
      const float4 v01 = tb[y0 * W_ + x1];
      const float4 v10 = tb[y1 * W_ + x0];
      const float4 v11 = tb[y1 * W_ + x1];
      acc0 = fmaf(w00, v00.x, fmaf(w01, v01.x, fmaf(w10, v10.x, fmaf(w11, v11.x, acc0))));
      acc1 = fmaf(w00, v00.y, fmaf(w01, v01.y, fmaf(w10, v10.y, fmaf(w11, v11.y, acc1))));
      acc2 = fmaf(w00, v00.z, fmaf(w01, v01.z, fmaf(w10, v10.z, fmaf(w11, v11.z, acc2))));
    } else {
      const float* pb = img + (size_t)b * C_ * HW;
      const int i00 = y0 * W_ + x0, i01 = y0 * W_ + x1;
      const int i10 = y1 * W_ + x0, i11 = y1 * W_ + x1;
      const float* c0 = pb;
      const float* c1 = pb + HW;
      const float* c2 = pb + 2 * HW;
      acc0 = fmaf(w00, c0[i00], fmaf(w01, c0[i01], fmaf(w10, c0[i10], fmaf(w11, c0[i11], acc0))));
      acc1 = fmaf(w00, c1[i00], fmaf(w01, c1[i01], fmaf(w10, c1[i10], fmaf(w11, c1[i11], acc1))));
      acc2 = fmaf(w00, c2[i00], fmaf(w01, c2[i01], fmaf(w10, c2[i10], fmaf(w11, c2[i11], acc2))));
    }
    buf ^= 1;
  }

  float* ob = out + (size_t)b * C_ * HW + p;
  ob[0] = acc0;
  ob[HW] = acc1;
  ob[2 * HW] = acc2;
}

extern "C" void kernel_launch(void* const* d_in, const int* in_sizes, int n_in,
                              void* d_out, int out_size, void* d_ws,
                              size_t ws_size, hipStream_t stream) {
  (void)in_sizes; (void)n_in; (void)out_size;
  const float* image = (const float*)d_in[0];
  const float* offsets = (const float*)d_in[1];
  const float* weights = (const float*)d_in[2];
  float* out = (float*)d_out;

  const int nblocks = (B_ * HW) / BLOCK;  // 1024
  const size_t need = (size_t)B_ * HW * sizeof(float4);  // 4 MB

  if (d_ws != nullptr && ws_size >= need) {
    float4* timg = (float4*)d_ws;
    nchw_to_nhwc4_kernel<<<nblocks, BLOCK, 0, stream>>>(image, timg);
    adacof_kernel<true><<<nblocks, BLOCK, 0, stream>>>(image, offsets, weights,
                                                       timg, out);
  } else {
    adacof_kernel<false><<<nblocks, BLOCK, 0, stream>>>(image, offsets, weights,
                                                        nullptr, out);
  }
}